// CARFACCell_11819749998690
// MI455X (gfx1250) — compile-verified
//
#include <hip/hip_runtime.h>
#include <hip/hip_bf16.h>
#include <stdint.h>

#define T_LEN 4096
#define BLOCK 256
#define PER_T 16

typedef float v2f __attribute__((ext_vector_type(2)));
typedef float v4f __attribute__((ext_vector_type(4)));
typedef float v8f __attribute__((ext_vector_type(8)));
typedef int   v4i __attribute__((vector_size(4 * sizeof(int))));

#define AS1 __attribute__((address_space(1)))
#define AS3 __attribute__((address_space(3)))

#if defined(__HIP_DEVICE_COMPILE__)
# if __has_builtin(__builtin_amdgcn_global_load_async_to_lds_b128) && __has_builtin(__builtin_amdgcn_s_wait_asynccnt)
#  define USE_ASYNC_LD 1
# else
#  define USE_ASYNC_LD 0
# endif
# if __has_builtin(__builtin_amdgcn_global_store_async_from_lds_b128) && __has_builtin(__builtin_amdgcn_s_wait_asynccnt)
#  define USE_ASYNC_ST 1
# else
#  define USE_ASYNC_ST 0
# endif
# if __has_builtin(__builtin_amdgcn_wmma_f32_16x16x4_f32)
#  define USE_WMMA 1
# else
#  define USE_WMMA 0
# endif
#else
# define USE_ASYNC_LD 0
# define USE_ASYNC_ST 0
# define USE_WMMA 0
#endif

// symmetric padding index map: -1->0, -2->1, T->T-1, T+1->T-2 (jnp.pad 'symmetric')
__device__ __forceinline__ int symi(int i) {
    i = (i < 0) ? (-1 - i) : i;
    i = (i >= T_LEN) ? (2 * T_LEN - 1 - i) : i;
    return i;
}

__global__ __launch_bounds__(BLOCK)
void rec_smooth_kernel(const float* __restrict__ a0,
                       const float* __restrict__ f,
                       const float* __restrict__ g,
                       const float* __restrict__ kern,
                       const int*   __restrict__ steps,
                       float* __restrict__ out)
{
    __shared__ __align__(16) float sA[T_LEN];   // ping
    __shared__ __align__(16) float sB[T_LEN];   // pong
    __shared__ float sF[BLOCK];
    __shared__ float sG[BLOCK];

    const int row = blockIdx.x;
    const int tid = threadIdx.x;
    const size_t rowOff = (size_t)row * T_LEN;
    const float* fRow = f + rowOff;
    const float* gRow = g + rowOff;

    // ---------- stage f,g rows to LDS (CDNA5 async copy) ----------
#if USE_ASYNC_LD
    {
#pragma unroll
        for (int k = 0; k < 4; ++k) {
            const int e = k * 1024 + tid * 4;   // 16B per lane, fully coalesced
            __builtin_amdgcn_global_load_async_to_lds_b128(
                (AS1 v4i*)(fRow + e), (AS3 v4i*)(sA + e), 0, 0);
            __builtin_amdgcn_global_load_async_to_lds_b128(
                (AS1 v4i*)(gRow + e), (AS3 v4i*)(sB + e), 0, 0);
        }
        __builtin_amdgcn_s_wait_asynccnt(0);
    }
#else
    {
#pragma unroll
        for (int k = 0; k < 4; ++k) {
            const int e = k * 1024 + tid * 4;
            *(v4f*)(sA + e) = *(const v4f*)(fRow + e);
            *(v4f*)(sB + e) = *(const v4f*)(gRow + e);
        }
    }
#endif
    __syncthreads();

    // ---------- per-thread chunk: local (F,G) composition ----------
    float rf[PER_T], rg[PER_T];
#pragma unroll
    for (int j = 0; j < PER_T; j += 4) {
        *(v4f*)(rf + j) = *(const v4f*)(sA + tid * PER_T + j);
        *(v4f*)(rg + j) = *(const v4f*)(sB + tid * PER_T + j);
    }
    float Fl = 1.0f, Gl = 0.0f;
#pragma unroll
    for (int j = 0; j < PER_T; ++j) {
        Gl = rf[j] * Gl + rg[j];   // combine(acc, elem): G = f*G + g
        Fl = Fl * rf[j];           //                     F = F*f
    }
    sF[tid] = Fl;
    sG[tid] = Gl;
    __syncthreads();

    // ---------- Hillis-Steele inclusive scan over 256 (F,G) pairs ----------
#pragma unroll
    for (int off = 1; off < BLOCK; off <<= 1) {
        float pF = 1.0f, pG = 0.0f;
        const bool has = (tid >= off);
        if (has) { pF = sF[tid - off]; pG = sG[tid - off]; }
        __syncthreads();
        if (has) {
            const float cF = sF[tid], cG = sG[tid];
            sF[tid] = pF * cF;          // (fx*fy,
            sG[tid] = cF * pG + cG;     //  fy*gx + gy)
        }
        __syncthreads();
    }

    const float a0v = a0[row];
    float a = (tid == 0) ? a0v : (sF[tid - 1] * a0v + sG[tid - 1]);
#pragma unroll
    for (int j = 0; j < PER_T; ++j) {
        a = rf[j] * a + rg[j];
        rf[j] = a;
    }
#pragma unroll
    for (int j = 0; j < PER_T; j += 4)
        *(v4f*)(sA + tid * PER_T + j) = *(const v4f*)(rf + j);
    __syncthreads();

    // ---------- smoothing: `steps` symmetric-padded 5-tap passes ----------
    const int   nsteps = steps[0];
    const float w0 = kern[0], w1 = kern[1], w2 = kern[2], w3 = kern[3], w4 = kern[4];

    float* ping = sA;
    float* pong = sB;

#if USE_WMMA
    // Row viewed as 256x16 matrix X; one pass is Y = X * W with W banded:
    // W[i][j] = w[i-j+2] for |i-j|<=2. Split into 6 K=4 chunks with column
    // bases cb = {-4,0,4,8,12,16} (cb=-4 / cb=16 are the cross-tile corners).
    const int lane = tid & 31;
    const int half = lane >> 4;     // 0: lanes 0-15, 1: lanes 16-31
    const int n    = lane & 15;     // N column of B/C/D
    v2f bw[6];
#pragma unroll
    for (int c = 0; c < 6; ++c) {
        // B 4x16 layout (mirror of documented 16x4 A layout):
        //   v0: K=2*half, v1: K=2*half+1, N = lane&15
        const int i0 = (-4 + 4 * c) + 2 * half;
        const int d0 = i0 - n;
        const int d1 = i0 + 1 - n;
        bw[c].x = (d0 == -2) ? w0 : (d0 == -1) ? w1 : (d0 == 0) ? w2
                : (d0 ==  1) ? w3 : (d0 ==  2) ? w4 : 0.0f;
        bw[c].y = (d1 == -2) ? w0 : (d1 == -1) ? w1 : (d1 == 0) ? w2
                : (d1 ==  1) ? w3 : (d1 ==  2) ? w4 : 0.0f;
    }
    const int wav = tid >> 5;
    for (int s = 0; s < nsteps; ++s) {
#pragma unroll
        for (int tt = 0; tt < 2; ++tt) {
            const int tile = wav * 2 + tt;   // 8 waves x 2 = 16 tiles of 16 rows
            const int tb   = tile * 256;     // element base of this tile
            v8f acc = {};
#pragma unroll
            for (int c = 0; c < 6; ++c) {
                const int cb = -4 + 4 * c;
                // A 16x4 layout (ISA table): lane<16 -> {K0,K1}, lane>=16 -> {K2,K3}, M=lane&15
                const int ia = tb + (lane & 15) * 16 + cb + 2 * half;
                v2f av;
                av.x = ping[symi(ia)];
                av.y = ping[symi(ia + 1)];
                acc = __builtin_amdgcn_wmma_f32_16x16x4_f32(
                          false, av, false, bw[c], (short)0, acc, false, false);
            }
            // D 16x16 f32 layout: VGPR v -> M = v + 8*half, N = lane&15
#pragma unroll
            for (int v = 0; v < 8; ++v) {
                const int m = v + 8 * half;
                pong[tb + m * 16 + n] = acc[v];
            }
        }
        __syncthreads();
        float* t = ping; ping = pong; pong = t;
    }
#else
    for (int s = 0; s < nsteps; ++s) {
#pragma unroll
        for (int j = 0; j < PER_T; ++j) {
            const int t = tid * PER_T + j;
            const float y = w0 * ping[symi(t - 2)] + w1 * ping[symi(t - 1)]
                          + w2 * ping[t]
                          + w3 * ping[symi(t + 1)] + w4 * ping[symi(t + 2)];
            pong[t] = y;
        }
        __syncthreads();
        float* t = ping; ping = pong; pong = t;
    }
#endif

    // ---------- write the row out ----------
    float* oRow = out + rowOff;
#if USE_ASYNC_ST
    {
#pragma unroll
        for (int k = 0; k < 4; ++k) {
            const int e = k * 1024 + tid * 4;
            __builtin_amdgcn_global_store_async_from_lds_b128(
                (AS1 v4i*)(oRow + e), (AS3 v4i*)(ping + e), 0, 0);
        }
        __builtin_amdgcn_s_wait_asynccnt(0);
    }
#else
    {
#pragma unroll
        for (int k = 0; k < 4; ++k) {
            const int e = k * 1024 + tid * 4;
            *(v4f*)(oRow + e) = *(const v4f*)(ping + e);
        }
    }
#endif
}

extern "C" void kernel_launch(void* const* d_in, const int* in_sizes, int n_in,
                              void* d_out, int out_size, void* d_ws, size_t ws_size,
                              hipStream_t stream) {
    (void)n_in; (void)out_size; (void)d_ws; (void)ws_size;
    const float* a0    = (const float*)d_in[0];
    const float* f     = (const float*)d_in[1];
    const float* g     = (const float*)d_in[2];
    const float* kr    = (const float*)d_in[3];
    const int*   steps = (const int*)d_in[4];
    float* out = (float*)d_out;
    const int rows = in_sizes[0];   // B*C = 4096 rows; in_sizes[1]/rows = T
    rec_smooth_kernel<<<rows, BLOCK, 0, stream>>>(a0, f, g, kr, steps, out);
}